// VGAEModel_10368051052753
// MI455X (gfx1250) — compile-verified
//
#include <hip/hip_runtime.h>
#include <hip/hip_bf16.h>

// ---------- problem constants ----------
constexpr int Bg   = 64;        // graphs
constexpr int NPG  = 1024;      // nodes per graph
constexpr int EPG  = 16384;     // edges per graph
constexpr int INF  = 64;        // input feature dim
constexpr int HID  = 128;
constexpr int OUTF = 32;
constexpr int Nn   = Bg * NPG;  // 65536 nodes
constexpr int Et   = Bg * EPG;  // 1048576 edges

typedef __attribute__((ext_vector_type(2))) float v2f;
typedef __attribute__((ext_vector_type(4))) float v4f;
typedef __attribute__((ext_vector_type(8))) float v8f;

// D = A(16x4,f32) * B(4x16,f32) + C(16x16,f32)
__device__ __forceinline__ v8f wmma4(v2f a, v2f b, v8f c) {
    return __builtin_amdgcn_wmma_f32_16x16x4_f32(
        /*neg_a=*/false, a, /*neg_b=*/false, b,
        /*c_mod=*/(short)0, c, /*reuse_a=*/false, /*reuse_b=*/false);
}

// ---------- degree accumulation (2M global f32 atomics) ----------
__global__ void k_degree(const int* __restrict__ src, const int* __restrict__ dst,
                         float* __restrict__ deg_out, float* __restrict__ deg_in) {
    int e = blockIdx.x * blockDim.x + threadIdx.x;
    if (e < Et) {
        atomicAdd(&deg_out[src[e]], 1.0f);
        atomicAdd(&deg_in [dst[e]], 1.0f);
    }
}

// deg -> rsqrt(max(deg,1)) in place
__global__ void k_rsqrt(float* __restrict__ rs_out, float* __restrict__ rs_in) {
    int i = blockIdx.x * blockDim.x + threadIdx.x;
    if (i < Nn) {
        rs_out[i] = rsqrtf(fmaxf(rs_out[i], 1.0f));
        rs_in [i] = rsqrtf(fmaxf(rs_in [i], 1.0f));
    }
}

// hs[n][c] = features[n][c] * rs_out[n]   (vector of 4 per thread)
__global__ void k_prescale(const float* __restrict__ feat, const float* __restrict__ rs_out,
                           float* __restrict__ hs) {
    int i = (blockIdx.x * blockDim.x + threadIdx.x) * 4;
    if (i < Nn * INF) {
        const float r = rs_out[i >> 6];             // 4 consecutive stay in one row
        v4f f = *(const v4f*)&feat[i];
        f.x *= r; f.y *= r; f.z *= r; f.w *= r;
        *(v4f*)&hs[i] = f;
    }
}

// ---------- gconv1 aggregation: one workgroup per graph, 256KB LDS tile ----------
// tile[1024][64] accumulated with ds_add_f32; output m1 = tile * rs_in
__global__ __launch_bounds__(1024) void k_agg1(const float* __restrict__ hs,
                                               const int* __restrict__ src,
                                               const int* __restrict__ dst,
                                               const float* __restrict__ rs_in,
                                               float* __restrict__ m1) {
    extern __shared__ float tile[];                 // NPG*INF floats = 256 KB
    const int g = blockIdx.x;
    for (int i = threadIdx.x * 4; i < NPG * INF; i += 4096) *(v4f*)&tile[i] = (v4f){};
    __syncthreads();
    const int wave = threadIdx.x >> 5;
    const int c2   = (threadIdx.x & 31) * 2;        // lane -> channels c2, c2+1
    const int ebase = g * EPG, nbase = g * NPG;
    for (int e = wave; e < EPG; e += 32) {          // each wave owns an edge; lanes = channels
        const int s = src[ebase + e];
        const int d = dst[ebase + e] - nbase;       // graph-local
        const v2f v = *(const v2f*)&hs[s * INF + c2];   // one b64 coalesced gather
        atomicAdd(&tile[d * INF + c2    ], v.x);        // banks 2c / 2c+1: conflict-free
        atomicAdd(&tile[d * INF + c2 + 1], v.y);
    }
    __syncthreads();
    for (int i = threadIdx.x * 4; i < NPG * INF; i += 4096) {
        const int row = nbase + (i >> 6);
        const float r = rs_in[row];
        v4f t = *(const v4f*)&tile[i];
        t.x *= r; t.y *= r; t.z *= r; t.w *= r;
        *(v4f*)&m1[row * INF + (i & 63)] = t;
    }
}

// ---------- gconv2/3 shared aggregation: 2 channel-halves per graph ----------
__global__ __launch_bounds__(1024) void k_agg2(const float* __restrict__ h1s,
                                               const int* __restrict__ src,
                                               const int* __restrict__ dst,
                                               const float* __restrict__ rs_in,
                                               float* __restrict__ m2) {
    extern __shared__ float tile[];                 // NPG*64 floats = 256 KB
    const int g  = blockIdx.x >> 1;
    const int c0 = (blockIdx.x & 1) * 64;
    for (int i = threadIdx.x * 4; i < NPG * 64; i += 4096) *(v4f*)&tile[i] = (v4f){};
    __syncthreads();
    const int wave = threadIdx.x >> 5;
    const int c2   = (threadIdx.x & 31) * 2;
    const int ebase = g * EPG, nbase = g * NPG;
    for (int e = wave; e < EPG; e += 32) {
        const int s = src[ebase + e];
        const int d = dst[ebase + e] - nbase;
        const v2f v = *(const v2f*)&h1s[s * HID + c0 + c2];
        atomicAdd(&tile[d * 64 + c2    ], v.x);
        atomicAdd(&tile[d * 64 + c2 + 1], v.y);
    }
    __syncthreads();
    for (int i = threadIdx.x * 4; i < NPG * 64; i += 4096) {
        const int row = nbase + (i >> 6);
        const float r = rs_in[row];
        v4f t = *(const v4f*)&tile[i];
        t.x *= r; t.y *= r; t.z *= r; t.w *= r;
        *(v4f*)&m2[row * HID + c0 + (i & 63)] = t;
    }
}

// ---------- gconv1 linear: h1s = relu(m1 @ W1 + b1) * rs_out  (fused next-layer prescale)
// tiles: (Nn/16) x (HID/16) = 4096 x 8; 8 waves/block, each wave one 16x16 tile
__global__ __launch_bounds__(256) void k_gemm1(const float* __restrict__ m1,
                                               const float* __restrict__ W1,
                                               const float* __restrict__ b1,
                                               const float* __restrict__ rs_out,
                                               float* __restrict__ h1s) {
    __shared__ float w[INF * HID];                  // 32 KB
    // CDNA5 async DMA: global -> LDS without VGPR round-trip (ASYNCcnt-tracked)
    for (int i = threadIdx.x * 4; i < INF * HID; i += 256 * 4) {
        const float*  gp  = &W1[i];
        unsigned      lp  = (unsigned)(size_t)&w[i];    // 32-bit LDS address
        asm volatile("global_load_async_to_lds_b128 %0, %1, off"
                     :: "v"(lp), "v"(gp) : "memory");
    }
    asm volatile("s_wait_asynccnt 0" ::: "memory");
    __syncthreads();

    const int wave = threadIdx.x >> 5, lane = threadIdx.x & 31;
    const int tile = blockIdx.x * 8 + wave;
    const int tm = tile >> 3, tn = tile & 7;
    const int mloc = lane & 15, half = lane >> 4;
    const int arow = tm * 16 + mloc;                // A: lane -> row M
    const int n    = tn * 16 + mloc;                // B/C: lane -> col N
    v8f acc = {};
    #pragma unroll
    for (int k = 0; k < INF; k += 4) {
        const int kk = k + 2 * half;
        v2f a = *(const v2f*)&m1[arow * INF + kk];
        v2f b;
        b.x = w[kk * HID + n]; b.y = w[(kk + 1) * HID + n];
        acc = wmma4(a, b, acc);
    }
    const float bias = b1[n];
    #pragma unroll
    for (int r = 0; r < 8; ++r) {
        const int orow = tm * 16 + r + 8 * half;
        float v = acc[r] + bias;
        v = v > 0.0f ? v : 0.0f;                    // relu
        h1s[orow * HID + n] = v * rs_out[orow];     // prescale for next gather
    }
}

// ---------- dual linear + reparameterization:
// mean = m2@W2+b2 ; log_std = m2@W3+b3 ; z = mean + noise*exp(log_std)
// tiles: (Nn/16) x (OUTF/16) = 4096 x 2
__global__ __launch_bounds__(256) void k_gemm23(const float* __restrict__ m2,
                                                const float* __restrict__ W2, const float* __restrict__ b2,
                                                const float* __restrict__ W3, const float* __restrict__ b3,
                                                const float* __restrict__ noise,
                                                float* __restrict__ meanO, float* __restrict__ lsO,
                                                float* __restrict__ z) {
    __shared__ float w2[HID * OUTF];                // 16 KB
    __shared__ float w3[HID * OUTF];                // 16 KB
    for (int i = threadIdx.x; i < HID * OUTF; i += 256) { w2[i] = W2[i]; w3[i] = W3[i]; }
    __syncthreads();
    const int wave = threadIdx.x >> 5, lane = threadIdx.x & 31;
    const int tile = blockIdx.x * 8 + wave;
    const int tm = tile >> 1, tn = tile & 1;
    const int mloc = lane & 15, half = lane >> 4;
    const int arow = tm * 16 + mloc;
    const int n    = tn * 16 + mloc;
    v8f accM = {}, accL = {};
    #pragma unroll
    for (int k = 0; k < HID; k += 4) {
        const int kk = k + 2 * half;
        v2f a = *(const v2f*)&m2[arow * HID + kk];
        v2f bm, bl;
        bm.x = w2[kk * OUTF + n]; bm.y = w2[(kk + 1) * OUTF + n];
        bl.x = w3[kk * OUTF + n]; bl.y = w3[(kk + 1) * OUTF + n];
        accM = wmma4(a, bm, accM);
        accL = wmma4(a, bl, accL);
    }
    const float biasM = b2[n], biasL = b3[n];
    #pragma unroll
    for (int r = 0; r < 8; ++r) {
        const int orow = tm * 16 + r + 8 * half;
        const int idx = orow * OUTF + n;
        const float mv = accM[r] + biasM;
        const float lv = accL[r] + biasL;
        meanO[idx] = mv;
        lsO[idx]  = lv;
        z[idx]    = mv + noise[idx] * __expf(lv);
    }
}

// ---------- adjacency reconstruction: per graph sigmoid(z z^T), NT stores ----------
// Each wave computes a 32(M) x 64(N) patch: A (2 row-tiles) preloaded to regs,
// each B fragment loaded once and fed to two WMMAs -> load traffic ~1.5KB per 1KB stored.
// jobs: 64 graphs * 32 row-pairs * 16 col-groups = 32768 ; 8 waves/block -> 4096 blocks
__global__ __launch_bounds__(256) void k_adj(const float* __restrict__ z,
                                             float* __restrict__ adj) {
    const int wave = threadIdx.x >> 5, lane = threadIdx.x & 31;
    const int jid = blockIdx.x * 8 + wave;
    const int g   = jid >> 9;                       // 512 jobs per graph
    const int r0  = jid & 511;
    const int tig = r0 >> 4;                        // 0..31 (covers rows tig*32 .. +31)
    const int tjg = r0 & 15;                        // 0..15 (covers cols tjg*64 .. +63)
    const int mloc = lane & 15, half = lane >> 4;
    const float* zg = z + (long)g * NPG * OUTF;

    const int arow0 = tig * 32 + mloc;
    const int arow1 = arow0 + 16;
    v2f a0[8], a1[8];
    #pragma unroll
    for (int kk8 = 0; kk8 < 8; ++kk8) {             // K = 32 = 8 steps of 4
        const int kk = kk8 * 4 + 2 * half;
        a0[kk8] = *(const v2f*)&zg[arow0 * OUTF + kk];
        a1[kk8] = *(const v2f*)&zg[arow1 * OUTF + kk];
    }

    float* out = adj + (long)g * NPG * NPG;
    #pragma unroll
    for (int j = 0; j < 4; ++j) {
        const int tj   = tjg * 4 + j;
        const int bcol = tj * 16 + mloc;            // B[k][n] = z[n][k]
        v8f acc0 = {}, acc1 = {};
        #pragma unroll
        for (int kk8 = 0; kk8 < 8; ++kk8) {
            const int kk = kk8 * 4 + 2 * half;
            v2f b = *(const v2f*)&zg[bcol * OUTF + kk];
            acc0 = wmma4(a0[kk8], b, acc0);
            acc1 = wmma4(a1[kk8], b, acc1);
        }
        #pragma unroll
        for (int r = 0; r < 8; ++r) {
            const int row0 = tig * 32 + r + 8 * half;
            const int col  = tj * 16 + mloc;
            const float s0 = 1.0f / (1.0f + __expf(-acc0[r]));
            const float s1 = 1.0f / (1.0f + __expf(-acc1[r]));
            __builtin_nontemporal_store(s0, &out[(long)row0 * NPG + col]);
            __builtin_nontemporal_store(s1, &out[(long)(row0 + 16) * NPG + col]);
        }
    }
}

// =========================== host launcher ===========================
extern "C" void kernel_launch(void* const* d_in, const int* in_sizes, int n_in,
                              void* d_out, int out_size, void* d_ws, size_t ws_size,
                              hipStream_t stream) {
    const float* feat  = (const float*)d_in[0];   // N x 64
    const int*   src   = (const int*)  d_in[1];   // Et
    const int*   dst   = (const int*)  d_in[2];   // Et
    const float* noise = (const float*)d_in[3];   // N x 32
    const float* W1    = (const float*)d_in[4];
    const float* b1    = (const float*)d_in[5];
    const float* W2    = (const float*)d_in[6];
    const float* b2    = (const float*)d_in[7];
    const float* W3    = (const float*)d_in[8];
    const float* b3    = (const float*)d_in[9];

    // workspace layout (floats)
    char* ws = (char*)d_ws;
    float* rs_out = (float*)ws;                         ws += (size_t)Nn * 4;        // 256 KB
    float* rs_in  = (float*)ws;                         ws += (size_t)Nn * 4;        // 256 KB
    char*  regA   = ws;                                 ws += (size_t)Nn * HID * 4;  // 32 MB (hs+m1, later m2)
    float* hs  = (float*)regA;                          // N x 64 (prescaled features)
    float* m1  = (float*)(regA + (size_t)Nn * INF * 4); // N x 64 (gconv1 aggregate)
    float* m2  = (float*)regA;                          // N x 128 (reuses hs+m1 after gemm1)
    float* h1s = (float*)ws;                            ws += (size_t)Nn * HID * 4;  // 32 MB
    float* zbuf= (float*)ws;                            ws += (size_t)Nn * OUTF * 4; // 8 MB

    // output layout: adj (64*1024*1024) | mean (N*32) | log_std (N*32)
    float* adj   = (float*)d_out;
    float* meanO = adj   + (size_t)Bg * NPG * NPG;
    float* lsO   = meanO + (size_t)Nn * OUTF;

    // allow 256 KB dynamic LDS for the per-graph aggregation kernels
    (void)hipFuncSetAttribute((const void*)k_agg1,
                              hipFuncAttributeMaxDynamicSharedMemorySize, NPG * INF * 4);
    (void)hipFuncSetAttribute((const void*)k_agg2,
                              hipFuncAttributeMaxDynamicSharedMemorySize, NPG * 64 * 4);

    // 1) degrees
    hipMemsetAsync(rs_out, 0, (size_t)Nn * 4, stream);
    hipMemsetAsync(rs_in,  0, (size_t)Nn * 4, stream);
    k_degree<<<Et / 256, 256, 0, stream>>>(src, dst, rs_out, rs_in);
    k_rsqrt <<<Nn / 256, 256, 0, stream>>>(rs_out, rs_in);

    // 2) gconv1: prescale -> LDS scatter-aggregate -> WMMA linear (+relu, +next prescale)
    k_prescale<<<(Nn * INF / 4) / 256, 256, 0, stream>>>(feat, rs_out, hs);
    k_agg1 <<<Bg, 1024, NPG * INF * 4, stream>>>(hs, src, dst, rs_in, m1);
    k_gemm1<<<(Nn / 16) * (HID / 16) / 8, 256, 0, stream>>>(m1, W1, b1, rs_out, h1s);

    // 3) shared aggregation for gconv2/gconv3 (two 64-channel passes per graph)
    k_agg2 <<<Bg * 2, 1024, NPG * 64 * 4, stream>>>(h1s, src, dst, rs_in, m2);

    // 4) mean / log_std / z in one fused dual-WMMA kernel
    k_gemm23<<<(Nn / 16) * (OUTF / 16) / 8, 256, 0, stream>>>(m2, W2, b2, W3, b3,
                                                              noise, meanO, lsO, zbuf);

    // 5) adjacency: 2x4 register-blocked f32 WMMA + sigmoid + NT streaming stores
    k_adj<<<(Bg * 32 * 16) / 8, 256, 0, stream>>>(zbuf, adj);
}